// DepthToSpace_472446402854
// MI455X (gfx1250) — compile-verified
//
#include <hip/hip_runtime.h>
#include <hip/hip_bf16.h>

// DepthToSpace (DCR), block_size = 2
//   input : float32 [B=16, D=256, H=128, W=128]
//   output: float32 [B=16, C=64, H2=256, W2=256]
//   out[b, c, h*2 + j, w*2 + r] = in[b, (j*2 + r)*C + c, h, w]
//
// Pure streaming permutation: 256 MiB in + 256 MiB out -> bandwidth-bound,
// roofline ~22 us @ 23.3 TB/s. One thread = 8 output floats:
//   two global_load_b128 (input rows d0 and d0+C, w0..w0+3, NT-hinted)
//   two global_store_b128 (interleaved, NT-hinted)
// All streams lane-contiguous; NT keeps the one-touch 512 MiB working set
// from thrashing the 192 MB L2.

typedef float v4f __attribute__((ext_vector_type(4)));

namespace {
constexpr int kD  = 256;               // input channels
constexpr int kH  = 128;
constexpr int kW  = 128;
constexpr int kBS = 2;
constexpr int kC  = kD / (kBS * kBS);  // 64 output channels
constexpr int kH2 = kH * kBS;          // 256
constexpr int kW2 = kW * kBS;          // 256
}

__global__ __launch_bounds__(256) void DepthToSpace_472446402854_kernel(
    const float* __restrict__ in, float* __restrict__ out, long long n_vec8) {
  const long long tid = (long long)blockIdx.x * blockDim.x + threadIdx.x;
  if (tid >= n_vec8) return;

  // tid = ((b*C + c)*H2 + oh) * (W2/8) + ow8
  const int ow8 = (int)(tid & (kW2 / 8 - 1));       // 0..31
  long long t = tid >> 5;
  const int oh = (int)(t & (kH2 - 1));              // 0..255
  t >>= 8;
  const int c = (int)(t & (kC - 1));                // 0..63
  const int b = (int)(t >> 6);

  const int j  = oh & 1;
  const int h  = oh >> 1;
  const int w0 = ow8 * 4;                           // input w for the 4 (r=0) lanes

  const int d0 = (j * kBS) * kC + c;                // r=0 row; r=1 row is d0 + C

  const long long inBase =
      (((long long)(b * kD + d0) * kH) + h) * kW + w0;
  const long long rowStride = (long long)kC * kH * kW;  // d -> d + C

  const v4f a  = __builtin_nontemporal_load((const v4f*)(in + inBase));              // row d0: w0..w0+3
  const v4f bv = __builtin_nontemporal_load((const v4f*)(in + inBase + rowStride));  // row d1: w0..w0+3

  v4f o0, o1;
  o0.x = a.x;   // ow+0 : r=0, w0
  o0.y = bv.x;  // ow+1 : r=1, w0
  o0.z = a.y;   // ow+2 : r=0, w0+1
  o0.w = bv.y;  // ow+3 : r=1, w0+1
  o1.x = a.z;   // ow+4 : r=0, w0+2
  o1.y = bv.z;  // ow+5 : r=1, w0+2
  o1.z = a.w;   // ow+6 : r=0, w0+3
  o1.w = bv.w;  // ow+7 : r=1, w0+3

  float* outp = out + tid * 8;
  __builtin_nontemporal_store(o0, (v4f*)outp);
  __builtin_nontemporal_store(o1, (v4f*)(outp + 4));
}

extern "C" void kernel_launch(void* const* d_in, const int* in_sizes, int n_in,
                              void* d_out, int out_size, void* d_ws, size_t ws_size,
                              hipStream_t stream) {
  (void)n_in; (void)d_ws; (void)ws_size; (void)out_size;
  const float* in = (const float*)d_in[0];
  float* out = (float*)d_out;

  const long long total = (long long)in_sizes[0];   // == out element count
  const long long n_vec8 = total / 8;               // one 8-float tile per thread

  const int block = 256;
  const long long grid = (n_vec8 + block - 1) / block;

  DepthToSpace_472446402854_kernel<<<(unsigned)grid, block, 0, stream>>>(in, out, n_vec8);
}